// ALSTM_B_46952582480192
// MI455X (gfx1250) — compile-verified
//
#include <hip/hip_runtime.h>
#include <math.h>

// ACT-LSTM forward for MI455X (gfx1250, wave32).
// GEMVs mapped onto V_WMMA_F32_16X16X4_F32: 16-row tiles, B = h-chunk
// replicated across the 16 N columns; column 0 (lanes 0/16) of D holds the
// 16 dot products. Memory-bound: W_hh (64MB fp32) is L2-resident (192MB L2).

typedef __attribute__((ext_vector_type(2))) float v2f;
typedef __attribute__((ext_vector_type(8))) float v8f;

constexpr int kI = 1024;   // input dim
constexpr int kH = 2048;   // hidden dim
constexpr int kO = 1024;   // output dim
constexpr int kG = 4 * kH; // 8192 gate rows
constexpr int kT = 20;     // MAX_STEPS
#define ACT_THRESH 0.99f   // 1 - EPS

// ---- 16-row GEMV tile via WMMA f32 16x16x4 ------------------------------
// Wb = W + (size_t)(r0 + (lane&15)) * ld   (per-lane row base)
// shh = h vector in LDS. Returns v8f: lane {0,16} column-0 values are
// y[r0 + (lane>>4)*8 + v].
__device__ __forceinline__ v8f gemv16_tile(const float* __restrict__ Wb,
                                           const float* __restrict__ shh,
                                           int K, int lane) {
  v8f acc0 = {0.f, 0.f, 0.f, 0.f, 0.f, 0.f, 0.f, 0.f};
  v8f acc1 = {0.f, 0.f, 0.f, 0.f, 0.f, 0.f, 0.f, 0.f};
  const int hi = lane >> 4;                 // 0: K 0..1 ; 1: K 2..3
  const float* wp = Wb + 2 * hi;            // A operand column offset
  const float* hp = shh + hi;               // B operand row offset
  #pragma unroll 2
  for (int k = 0; k < K; k += 8) {
    // chunk k..k+3
    v2f a0 = *(const v2f*)(wp + k);         // 8B-aligned: even offset
    v2f b0; b0.x = hp[k];     b0.y = hp[k + 2];
    acc0 = __builtin_amdgcn_wmma_f32_16x16x4_f32(
        false, a0, false, b0, (short)0, acc0, false, false);
    // chunk k+4..k+7 (second accumulator hides WMMA C-chain latency)
    v2f a1 = *(const v2f*)(wp + k + 4);
    v2f b1; b1.x = hp[k + 4]; b1.y = hp[k + 6];
    acc1 = __builtin_amdgcn_wmma_f32_16x16x4_f32(
        false, a1, false, b1, (short)0, acc1, false, false);
  }
  return acc0 + acc1;
}

__device__ __forceinline__ float sigmoidf_(float x) {
  return 1.f / (1.f + expf(-x));
}

// ---- one-time input projection: gx = W_ih[:,1:] @ x + b_ih + b_hh -------
// Also re-initializes the per-call scratch scalars (ws is poisoned 0xAA).
__global__ void __launch_bounds__(256)
act_gx_kernel(const float* __restrict__ W_ih, const float* __restrict__ x,
              const float* __restrict__ b_ih, const float* __restrict__ b_hh,
              float* __restrict__ gx, float* __restrict__ cum,
              int* __restrict__ done) {
  __shared__ float shx[kI];
  const int tid = threadIdx.x;
  #pragma unroll
  for (int q = 0; q < kI / 256; ++q) shx[tid + 256 * q] = x[tid + 256 * q];
  __syncthreads();
  const int r = blockIdx.x * 256 + tid;
  const float* wp = W_ih + (size_t)r * (size_t)(kI + 1) + 1;
  float s = b_ih[r] + b_hh[r];
  #pragma unroll 4
  for (int k = 0; k < kI; ++k) s += wp[k] * shx[k];
  gx[r] = s;
  if (blockIdx.x == 0 && tid == 0) { *cum = 0.f; *done = 0; }
}

// ---- one LSTM step: gates = W_hh@h + gx (+flag col) ; elementwise update -
// Block: 256 threads = 8 waves; block handles h-indices [32*b, 32*b+32).
// Wave w -> gate g=w>>1, rows g*2048 + j0 + (w&1)*16 .. +15.
__global__ void __launch_bounds__(256)
act_step_kernel(const float* __restrict__ W_hh, const float* __restrict__ gx,
                const float* __restrict__ W_ih,  // flag column (stride kI+1)
                const float* __restrict__ h_in, const float* __restrict__ c_in,
                float* __restrict__ h_out, float* __restrict__ c_out,
                const int* __restrict__ done, int first) {
  if (*done) return;  // uniform early-exit once halting step is selected
  __shared__ float shh[kH];
  __shared__ float shg[128];
  const int tid = threadIdx.x;
  #pragma unroll
  for (int q = 0; q < kH / 256; ++q) shh[tid + 256 * q] = h_in[tid + 256 * q];
  __syncthreads();

  const int lane = tid & 31;
  const int w = tid >> 5;
  const int g = w >> 1;
  const int j0 = blockIdx.x * 32;
  const int r0 = g * kH + j0 + (w & 1) * 16;
  const float* Wb = W_hh + (size_t)(r0 + (lane & 15)) * (size_t)kH;

  v8f acc = gemv16_tile(Wb, shh, kH, lane);

  if ((lane & 15) == 0) {  // column N=0 of D holds the 16 dot products
    const int base = w * 16 + (lane >> 4) * 8;
    #pragma unroll
    for (int v = 0; v < 8; ++v) shg[base + v] = acc[v];
  }
  __syncthreads();

  if (tid < 32) {  // fused LSTM elementwise update for this block's 32 j's
    const int j = j0 + tid;
    float gi = shg[tid]      + gx[j];
    float gf = shg[32 + tid] + gx[kH + j];
    float gg = shg[64 + tid] + gx[2 * kH + j];
    float go = shg[96 + tid] + gx[3 * kH + j];
    if (first) {  // step 0 sees input [1, x] -> add W_ih[:,0]
      gi += W_ih[(size_t)j * (kI + 1)];
      gf += W_ih[(size_t)(kH + j) * (kI + 1)];
      gg += W_ih[(size_t)(2 * kH + j) * (kI + 1)];
      go += W_ih[(size_t)(3 * kH + j) * (kI + 1)];
    }
    const float si = sigmoidf_(gi);
    const float sf = sigmoidf_(gf);
    const float so = sigmoidf_(go);
    const float cn = sf * c_in[j] + si * tanhf(gg);
    c_out[j] = cn;
    h_out[j] = so * tanhf(cn);
  }
}

// ---- halting probability + first-halt selection --------------------------
// Single workgroup. Writes h_out/c_out/ponder into d_out on the (unique)
// selected step; sets done so later steps early-exit.
__global__ void __launch_bounds__(256)
act_halt_kernel(const float* __restrict__ h_new, const float* __restrict__ c_new,
                const float* __restrict__ w_halt, const float* __restrict__ b_halt,
                float* __restrict__ cum, int* __restrict__ done,
                float* __restrict__ out, int t) {
  if (*done) return;
  __shared__ float red[256];
  __shared__ int ssel;
  const int tid = threadIdx.x;
  float s = 0.f;
  #pragma unroll
  for (int q = 0; q < kH / 256; ++q) {
    const int j = tid + 256 * q;
    s += w_halt[j] * h_new[j];
  }
  red[tid] = s;
  __syncthreads();
  for (int off = 128; off > 0; off >>= 1) {
    if (tid < off) red[tid] += red[tid + off];
    __syncthreads();
  }
  if (tid == 0) {
    const float p = sigmoidf_(red[0] + b_halt[0]);
    const float cn = *cum + p;
    *cum = cn;
    const int sel = (cn >= ACT_THRESH) || (t == kT - 1);  // forced at cap
    ssel = sel;
    if (sel) { *done = 1; out[kO + 2 * kH] = (float)t; }  // ponder = t*
  }
  __syncthreads();
  if (ssel) {
    #pragma unroll
    for (int q = 0; q < kH / 256; ++q) {
      const int j = tid + 256 * q;
      out[kO + j]      = h_new[j];   // h_out
      out[kO + kH + j] = c_new[j];   // c_out
    }
  }
}

// ---- final projection: out = W_out @ h_sel + b_out (WMMA tiles) ----------
__global__ void __launch_bounds__(256)
act_out_kernel(const float* __restrict__ W_out, const float* __restrict__ b_out,
               float* __restrict__ dout) {
  __shared__ float shh[kH];
  const int tid = threadIdx.x;
  const float* hsel = dout + kO;  // h_out written by the select kernel
  #pragma unroll
  for (int q = 0; q < kH / 256; ++q) shh[tid + 256 * q] = hsel[tid + 256 * q];
  __syncthreads();

  const int lane = tid & 31;
  const int w = tid >> 5;
  const int r0 = blockIdx.x * 128 + w * 16;
  const float* Wb = W_out + (size_t)(r0 + (lane & 15)) * (size_t)kH;

  v8f acc = gemv16_tile(Wb, shh, kH, lane);

  if ((lane & 15) == 0) {
    const int mbase = r0 + (lane >> 4) * 8;
    #pragma unroll
    for (int v = 0; v < 8; ++v) dout[mbase + v] = acc[v] + b_out[mbase + v];
  }
}

extern "C" void kernel_launch(void* const* d_in, const int* in_sizes, int n_in,
                              void* d_out, int out_size, void* d_ws, size_t ws_size,
                              hipStream_t stream) {
  const float* x      = (const float*)d_in[0];
  const float* h0     = (const float*)d_in[1];
  const float* c0     = (const float*)d_in[2];
  const float* W_ih   = (const float*)d_in[3];
  const float* W_hh   = (const float*)d_in[4];
  const float* b_ih   = (const float*)d_in[5];
  const float* b_hh   = (const float*)d_in[6];
  const float* w_halt = (const float*)d_in[7];
  const float* b_halt = (const float*)d_in[8];
  const float* W_out  = (const float*)d_in[9];
  const float* b_out  = (const float*)d_in[10];
  float* out = (float*)d_out;
  float* wsf = (float*)d_ws;

  // workspace layout (floats): gx[8192] | hA[2048] hB[2048] | cA[2048] cB[2048] | cum | done
  float* gx = wsf;
  float* hbuf[2] = {wsf + 8192, wsf + 8192 + 2048};
  float* cbuf[2] = {wsf + 8192 + 4096, wsf + 8192 + 4096 + 2048};
  float* cum = wsf + 16384;
  int* done  = (int*)(wsf + 16385);

  act_gx_kernel<<<kG / 256, 256, 0, stream>>>(W_ih, x, b_ih, b_hh, gx, cum, done);

  for (int t = 0; t < kT; ++t) {
    const float* hin = (t == 0) ? h0 : hbuf[t & 1];
    const float* cin = (t == 0) ? c0 : cbuf[t & 1];
    float* hout = hbuf[(t + 1) & 1];
    float* cout_ = cbuf[(t + 1) & 1];
    act_step_kernel<<<kH / 32, 256, 0, stream>>>(W_hh, gx, W_ih, hin, cin,
                                                 hout, cout_, done,
                                                 (t == 0) ? 1 : 0);
    act_halt_kernel<<<1, 256, 0, stream>>>(hout, cout_, w_halt, b_halt, cum,
                                           done, out, t);
  }

  act_out_kernel<<<kO / 128, 256, 0, stream>>>(W_out, b_out, out);
}